// MessagePassing_6828998000640
// MI455X (gfx1250) — compile-verified
//
#include <hip/hip_runtime.h>
#include <stdint.h>

#define CHUNK 1024
#define SCATTER_BLOCKS 512
#define THREADS 256

typedef unsigned int u32x4 __attribute__((ext_vector_type(4)));
typedef int          i32x8 __attribute__((ext_vector_type(8)));
typedef int          i32x4 __attribute__((ext_vector_type(4)));

typedef __attribute__((address_space(3))) unsigned char lds_byte;

// Recover the 32-bit LDS byte offset from a generic pointer to a __shared__ object.
__device__ __forceinline__ uint32_t lds_offset(const void* p) {
  return (uint32_t)(unsigned long long)(lds_byte*)p;
}

// Issue one TDM load: a 2-row tile (row-ids then col-ids, CHUNK int64 each)
// from the [2, E] edge_index tensor into LDS.  D# layout per CDNA5 ISA §8.3/8.4.
__device__ __forceinline__ void tdm_load_edges(const long long* gsrc,
                                               uint32_t lds_addr,
                                               uint32_t rem_elems,   // E - e0 (OOB -> zero fill)
                                               unsigned long long stride_elems) // = E
{
  unsigned long long ga = (unsigned long long)gsrc;

  u32x4 g0;
  g0.x = 1u;                                                   // count=1, no gather
  g0.y = lds_addr;                                             // lds_addr (bytes)
  g0.z = (uint32_t)ga;                                         // global_addr[31:0]
  g0.w = ((uint32_t)(ga >> 32) & 0x01FFFFFFu) | (2u << 30);    // global_addr[56:32] | type=2

  const uint32_t td0   = rem_elems;   // tensor_dim0 (elements remaining from tile start)
  const uint32_t td1   = 2u;          // tensor_dim1 (row array, col array)
  const uint32_t tile0 = CHUNK;       // tile_dim0
  const uint32_t tile1 = 2u;          // tile_dim1

  i32x8 g1;
  g1[0] = (int)(3u << 16);                                     // data_size=3 (8 bytes)
  g1[1] = (int)((td0 & 0xFFFFu) << 16);                        // tensor_dim0[15:0] @ bits 48..63
  g1[2] = (int)((td0 >> 16) | ((td1 & 0xFFFFu) << 16));        // tensor_dim0[31:16] | tensor_dim1[15:0]
  g1[3] = (int)((td1 >> 16) | (tile0 << 16));                  // tensor_dim1[31:16] | tile_dim0
  g1[4] = (int)(tile1 & 0xFFFFu);                              // tile_dim1 | tile_dim2=0
  g1[5] = (int)(uint32_t)(stride_elems & 0xFFFFFFFFu);         // tensor_dim0_stride[31:0]
  g1[6] = (int)(uint32_t)((stride_elems >> 32) & 0xFFFFu);     // stride[47:32] | dim1_stride lo = 0
  g1[7] = 0;

  i32x4 gz4 = {0, 0, 0, 0};                                    // groups 2/3 unused (2D tile)
  i32x8 gz8 = {0, 0, 0, 0, 0, 0, 0, 0};                        // extra group (6-arg toolchain form)
  __builtin_amdgcn_tensor_load_to_lds(g0, g1, gz4, gz4, gz8, 0);
}

__global__ void __launch_bounds__(THREADS)
scatter_kernel(const float* __restrict__ x,
               const long long* __restrict__ ei,
               float* __restrict__ out,
               long long E, int nchunks)
{
  __shared__ unsigned long long buf[2][2 * CHUNK];   // 32 KB: double-buffered {row[CHUNK], col[CHUNK]}
  const int tid = threadIdx.x;
  const int nb  = gridDim.x;

  long long c = blockIdx.x;
  if (c < (long long)nchunks && tid < 32) {          // wave 0 primes buffer 0
    const long long e0 = c * (long long)CHUNK;
    tdm_load_edges(ei + e0, lds_offset(&buf[0][0]),
                   (uint32_t)(E - e0), (unsigned long long)E);
  }

  int i = 0;
  for (; c < (long long)nchunks; c += nb, ++i) {
    const int cur = i & 1;
    const long long cnext = c + nb;

    if (tid < 32) {                                  // wave 0: prefetch next, drain current
      if (cnext < (long long)nchunks) {
        const long long e0n = cnext * (long long)CHUNK;
        tdm_load_edges(ei + e0n, lds_offset(&buf[cur ^ 1][0]),
                       (uint32_t)(E - e0n), (unsigned long long)E);
        __builtin_amdgcn_s_wait_tensorcnt(1);        // current chunk resident (in-order TDM)
      } else {
        __builtin_amdgcn_s_wait_tensorcnt(0);
      }
    }
    __syncthreads();                                 // publish LDS to all 8 waves

    const long long e0 = c * (long long)CHUNK;
    const long long remll = E - e0;
    const int count = (int)((remll < (long long)CHUNK) ? remll : (long long)CHUNK);
    const int sub4 = (tid & 7) * 4;                  // 8 lanes per edge, float4 each

    // Indices are int64 little-endian; node ids < 2^31, so read the low dword only.
    const unsigned int* idx32 = (const unsigned int*)&buf[cur][0];

#pragma unroll 2
    for (int li = tid >> 3; li < count; li += THREADS / 8) {
      const int r  = (int)idx32[2 * li];                       // row  (ds_load_b32)
      const int cc = (int)idx32[2 * (CHUNK + li)];             // col  (ds_load_b32)
      const float4 v = *(const float4*)(x + cc * 32 + sub4);   // global_load_b128, L2-resident
      float* o = out + r * 32 + sub4;
      unsafeAtomicAdd(o + 0, v.x);                   // global_atomic_add_f32 (no return)
      unsafeAtomicAdd(o + 1, v.y);
      unsafeAtomicAdd(o + 2, v.z);
      unsafeAtomicAdd(o + 3, v.w);
    }
    __syncthreads();                                 // buffer safe to overwrite next iteration
  }
}

__global__ void __launch_bounds__(THREADS)
zero_kernel(float4* __restrict__ out, int n4) {
  int i = blockIdx.x * blockDim.x + threadIdx.x;
  if (i < n4) out[i] = make_float4(0.f, 0.f, 0.f, 0.f);
}

extern "C" void kernel_launch(void* const* d_in, const int* in_sizes, int n_in,
                              void* d_out, int out_size, void* d_ws, size_t ws_size,
                              hipStream_t stream) {
  const float*     x   = (const float*)d_in[0];
  const long long* ei  = (const long long*)d_in[1];
  float*           out = (float*)d_out;

  const long long E = (long long)in_sizes[1] / 2;    // edge_index is [2, E] int64

  // 1) zero the output (atomics accumulate; harness poisons d_out)
  const int n4 = out_size / 4;
  hipLaunchKernelGGL(zero_kernel, dim3((n4 + THREADS - 1) / THREADS), dim3(THREADS),
                     0, stream, (float4*)out, n4);

  // 2) TDM-staged gather + atomic scatter
  const int nchunks = (int)((E + CHUNK - 1) / CHUNK);
  int blocks = SCATTER_BLOCKS;
  if (blocks > nchunks) blocks = nchunks;
  if (blocks < 1) blocks = 1;
  hipLaunchKernelGGL(scatter_kernel, dim3(blocks), dim3(THREADS),
                     0, stream, x, ei, out, E, nchunks);
}